// Quantizer_23519240913578
// MI455X (gfx1250) — compile-verified
//
#include <hip/hip_runtime.h>
#include <hip/hip_bf16.h>

// VQ-VAE quantizer for MI455X (gfx1250, wave32).
// score[m,n] = e_sq[m] + sum_k cb[m,k] * (-2*x[n,k]), accumulated with
// V_WMMA_F32_16X16X4_F32. C initialized with e_sq; B (x) pre-scaled by -2 and
// kept entirely in VGPRs; A (codebook) staged through LDS with double-buffered
// async copies (global_load_async_to_lds_b128 when available).

typedef float v2f __attribute__((ext_vector_type(2)));
typedef float v8f __attribute__((ext_vector_type(8)));
typedef int   v4i_gcc __attribute__((vector_size(16)));  // matches builtin proto

#define KCB   2048      // codebook size
#define DIM   256       // latent dim
#define BB    16        // batch
#define TT    2048      // time
#define BT    (BB * TT) // 32768 query vectors
#define NQUANT ((size_t)BB * DIM * TT)   // 8388608 floats of quant_out

#define CHUNK_ROWS 64                    // codebook rows staged per chunk
#define NCHUNK     (KCB / CHUNK_ROWS)    // 32
#define LDSPITCH   260                   // float pitch: bank-conflict-free, 16B-mult

#if defined(__has_builtin)
# if __has_builtin(__builtin_amdgcn_global_load_async_to_lds_b128)
#  define USE_ASYNC 1
# endif
#endif

typedef __attribute__((address_space(1))) v4i_gcc* as1_v4i;
typedef __attribute__((address_space(3))) v4i_gcc* as3_v4i;

__device__ __forceinline__ void wait_async_all() {
#ifdef USE_ASYNC
# if __has_builtin(__builtin_amdgcn_s_wait_asynccnt)
  __builtin_amdgcn_s_wait_asynccnt(0);
# else
  asm volatile("s_wait_asynccnt 0" ::: "memory");
# endif
#endif
}

// Block-wide copy of one 64x256 codebook chunk into padded LDS.
// 64 rows x 1024B: 4096 16-byte transfers, 16 per thread.
__device__ __forceinline__ void copy_chunk(const float* __restrict__ cb,
                                           float* __restrict__ dst,
                                           int chunk) {
  const float* src = cb + (size_t)chunk * CHUNK_ROWS * DIM;
  int tid = threadIdx.x;
  #pragma unroll
  for (int i = 0; i < 16; ++i) {
    int j = tid + 256 * i;          // 0..4095
    int row = j >> 6;               // 0..63
    int c4  = (j & 63) << 2;        // float column, multiple of 4
    const float* g = src + row * DIM + c4;
    float* l = dst + row * LDSPITCH + c4;
#ifdef USE_ASYNC
    __builtin_amdgcn_global_load_async_to_lds_b128(
        (as1_v4i)(uintptr_t)g,
        (as3_v4i)(uint32_t)(uintptr_t)l,
        /*offset=*/0, /*cpol=*/0);
#else
    *(float4*)l = *(const float4*)g;
#endif
  }
}

// ---------------------------------------------------------------------------
// K1: e_sq[k] = sum_c codebook[k][c]^2 ; zero the loss accumulator.
__global__ void vq_prep(const float* __restrict__ cb,
                        float* __restrict__ e_sq,
                        float* __restrict__ acc) {
  if (blockIdx.x == 0 && threadIdx.x == 0) *acc = 0.0f;
  int wave = (blockIdx.x * blockDim.x + threadIdx.x) >> 5;
  int lane = threadIdx.x & 31;
  if (wave >= KCB) return;
  const float* row = cb + (size_t)wave * DIM;
  float s = 0.0f;
  #pragma unroll
  for (int i = 0; i < DIM / 32; ++i) {
    float v = row[lane + 32 * i];
    s += v * v;
  }
  #pragma unroll
  for (int off = 16; off > 0; off >>= 1) s += __shfl_xor(s, off, 32);
  if (lane == 0) e_sq[wave] = s;
}

// ---------------------------------------------------------------------------
// K2: WMMA distance + argmin.
// Block = 8 waves; each wave owns a 16-wide N-tile of queries (block covers
// 128 consecutive bt). The codebook is swept in 32 double-buffered 64-row
// chunks staged in LDS; each chunk yields 4 M-tiles x 64 K-steps of WMMA.
//
// A layout (f32 16x4): lanes 0-15 -> M=lane, K=k0+{0,1}; lanes 16-31 -> K=k0+{2,3}.
// B layout (f32 4x16): mirrored — reg0/reg1 hold K=k0+2*half+{0,1}, N=lane%16.
// C layout: vgpr r, half h -> m = m0 + 8*h + r, n = lane%16.
__global__ void __launch_bounds__(256)
vq_argmin(const float* __restrict__ x,
          const float* __restrict__ cb,
          const float* __restrict__ e_sq,
          int* __restrict__ idx_ws,
          float* __restrict__ idx_out) {
  __shared__ __align__(16) float cbuf[2][CHUNK_ROWS * LDSPITCH];  // 133 KB

  int lane = threadIdx.x & 31;
  int wid  = threadIdx.x >> 5;                 // wave in block, 0..7
  int lm = lane & 15;                          // N (query) / M (cb row) in tile
  int hl = lane >> 4;                          // half-wave select

  int q0 = blockIdx.x * 128 + wid * 16;        // first query of this wave
  int b  = q0 / TT;
  int t0 = q0 % TT;
  const float* xb = x + (size_t)b * DIM * TT;
  int toff = t0 + lm;

  // Preload this wave's entire B operand (K=256 x N=16) into 128 VGPRs,
  // folding the -2 of the distance formula in once.
  v2f bfrag[64];
  #pragma unroll
  for (int k = 0; k < 64; ++k) {
    int ka = 4 * k + 2 * hl;
    v2f bv;
    bv.x = xb[(size_t)ka * TT + toff];
    bv.y = xb[(size_t)(ka + 1) * TT + toff];
    bfrag[k] = bv * -2.0f;
  }

  float bestv = 3.4e38f;
  int   besti = 0;

  copy_chunk(cb, &cbuf[0][0], 0);
  for (int cc = 0; cc < NCHUNK; ++cc) {
    wait_async_all();          // this wave's async copies complete
    __syncthreads();           // all waves' copies visible; prev buffer free
    if (cc + 1 < NCHUNK) copy_chunk(cb, &cbuf[(cc + 1) & 1][0], cc + 1);

    const float* bufp = &cbuf[cc & 1][0];
    #pragma unroll
    for (int mt = 0; mt < 4; ++mt) {
      int m0 = cc * CHUNK_ROWS + mt * 16;
      v8f acc = *(const v8f*)(e_sq + m0 + 8 * hl);   // C init = e_sq[m]
      const v2f* ap = (const v2f*)(bufp + (mt * 16 + lm) * LDSPITCH + 2 * hl);
      #pragma unroll
      for (int k = 0; k < 64; ++k) {
        v2f a = ap[2 * k];     // ds_load_b64, bank-conflict-free (pitch 260)
        acc = __builtin_amdgcn_wmma_f32_16x16x4_f32(
            false, a, false, bfrag[k], (short)0, acc, false, false);
      }
      // per-lane min over the 8 rows this lane holds (ascending m)
      float tv = acc[0];
      int   ti = m0 + 8 * hl;
      #pragma unroll
      for (int r = 1; r < 8; ++r) {
        float v = acc[r];
        int   m = m0 + 8 * hl + r;
        if (v < tv) { tv = v; ti = m; }
      }
      // merge with the other half-wave (same n, other 8 rows)
      float ov = __shfl_xor(tv, 16, 32);
      int   oi = __shfl_xor(ti, 16, 32);
      if (ov < tv || (ov == tv && oi < ti)) { tv = ov; ti = oi; }
      // fold tile result into running best (ties -> smaller index)
      if (tv < bestv || (tv == bestv && ti < besti)) { bestv = tv; besti = ti; }
    }
  }

  if (lane < 16) {
    int bt = q0 + lm;
    idx_ws[bt]  = besti;
    idx_out[bt] = (float)besti;   // d_out idx section (float dtype buffer)
  }
}

// ---------------------------------------------------------------------------
// K3: gather codebook[idx] -> quant_out in (B,C,T) layout via padded LDS
// transpose; accumulate sum((quant - x)^2) for both losses.
__global__ void vq_gather(const float* __restrict__ x,
                          const float* __restrict__ cb,
                          const int* __restrict__ idx_ws,
                          float* __restrict__ quant,
                          float* __restrict__ acc) {
  __shared__ float lds[16 * 257];  // 257 stride kills bank conflicts
  __shared__ int   rows[16];
  int tt  = blockIdx.x;            // 0..2047, tile of 16 bt values
  int tid = threadIdx.x;
  int b   = (tt * 16) / TT;
  int t0  = (tt * 16) % TT;

  if (tid < 16) rows[tid] = idx_ws[tt * 16 + tid];
  __syncthreads();

  // coalesced codebook row loads -> LDS (row n, col c)
  #pragma unroll
  for (int j = tid; j < 16 * DIM; j += 256) {
    int n = j >> 8, c = j & 255;
    lds[n * 257 + c] = cb[(size_t)rows[n] * DIM + c];
  }
  __syncthreads();

  // coalesced (B,C,T) stores: 16 consecutive t per c
  float s = 0.0f;
  #pragma unroll
  for (int j = tid; j < 16 * DIM; j += 256) {
    int c = j >> 4, n = j & 15;
    float q = lds[n * 257 + c];
    size_t o = ((size_t)b * DIM + c) * TT + t0 + n;
    float xv = x[o];
    quant[o] = q;            // quant_out == quant numerically (STE)
    float d = q - xv;
    s += d * d;
  }
  #pragma unroll
  for (int off = 16; off > 0; off >>= 1) s += __shfl_xor(s, off, 32);
  if ((tid & 31) == 0) atomicAdd(acc, s);
}

// ---------------------------------------------------------------------------
// K4: finalize the two scalar losses.
__global__ void vq_final(const float* __restrict__ acc,
                         float* __restrict__ losses) {
  float m = *acc / (float)NQUANT;
  losses[0] = m;           // codebook_loss = mean((quant - x)^2)
  losses[1] = 0.25f * m;   // commitment_loss = beta * same mean
}

// ---------------------------------------------------------------------------
extern "C" void kernel_launch(void* const* d_in, const int* in_sizes, int n_in,
                              void* d_out, int out_size, void* d_ws, size_t ws_size,
                              hipStream_t stream) {
  const float* x  = (const float*)d_in[0];   // (16, 256, 2048) fp32
  const float* cb = (const float*)d_in[1];   // (2048, 256) fp32
  float* out = (float*)d_out;
  // d_out layout (return order): quant_out | codebook_loss | commitment_loss | idx
  float* out_quant  = out;
  float* out_losses = out + NQUANT;
  float* out_idx    = out + NQUANT + 2;

  // workspace: e_sq (8KB) | acc (pad to 256B) | idx int32 (128KB)
  float* e_sq = (float*)d_ws;
  float* acc  = (float*)((char*)d_ws + 2048 * sizeof(float));
  int*   idxw = (int*)((char*)d_ws + 2048 * sizeof(float) + 256);

  vq_prep  <<<KCB / 8, 256, 0, stream>>>(cb, e_sq, acc);
  vq_argmin<<<BT / 128, 256, 0, stream>>>(x, cb, e_sq, idxw, out_idx);
  vq_gather<<<BT / 16, 256, 0, stream>>>(x, cb, idxw, out_quant, acc);
  vq_final <<<1, 1, 0, stream>>>(acc, out_losses);
}